// GCN_60610578482006
// MI455X (gfx1250) — compile-verified
//
#include <hip/hip_runtime.h>

#define N_NODES 10000
#define N_EDGES 160000
#define FDIM    512

typedef __attribute__((ext_vector_type(2))) float v2f;
typedef __attribute__((ext_vector_type(8))) float v8f;
typedef __attribute__((ext_vector_type(4))) unsigned int v4u;
typedef __attribute__((ext_vector_type(4))) int v4i;
typedef __attribute__((ext_vector_type(8))) int v8i;

// ---------------------------------------------------------------- utilities
__global__ void gcn_zero_kernel(float* __restrict__ p, int n) {
    int i = blockIdx.x * blockDim.x + threadIdx.x;
    int stride = gridDim.x * blockDim.x;
    for (; i < n; i += stride) p[i] = 0.0f;
}

// one thread per edge: count degrees with hardware f32 atomics
__global__ void gcn_degree_kernel(const int* __restrict__ src,
                                  const int* __restrict__ dst,
                                  float* __restrict__ deg_out,
                                  float* __restrict__ deg_in) {
    int e = blockIdx.x * blockDim.x + threadIdx.x;
    if (e < N_EDGES) {
        unsafeAtomicAdd(&deg_out[src[e]], 1.0f);
        unsafeAtomicAdd(&deg_in[dst[e]], 1.0f);
    }
}

// deg laid out as [deg_out | deg_in] (2*N contiguous) -> [norm_out | norm_in]
__global__ void gcn_norm_kernel(const float* __restrict__ deg,
                                float* __restrict__ norm, int n) {
    int i = blockIdx.x * blockDim.x + threadIdx.x;
    if (i < n) {
        float d = deg[i];
        norm[i] = (d > 0.0f) ? (1.0f / sqrtf(d)) : 0.0f;
    }
}

// ------------------------------------------------------------- aggregation
// agg[d] += x[s] * norm_out[s] * norm_in[d]   (norm_in folded in here so the
// GEMM's A tile is a pure 2D copy -> TDM-eligible)
// 128 threads cooperate on one edge; each thread moves 4 consecutive floats.
__global__ __launch_bounds__(256)
void gcn_aggregate_kernel(const float* __restrict__ x,
                          const int* __restrict__ src,
                          const int* __restrict__ dst,
                          const float* __restrict__ norm_out,
                          const float* __restrict__ norm_in,
                          float* __restrict__ agg) {
    int tid = threadIdx.x;
    int e = blockIdx.x * 2 + (tid >> 7);
    if (e >= N_EDGES) return;
    int f = (tid & 127) << 2;

    int s = src[e];
    int d = dst[e];
    float w = norm_out[s] * norm_in[d];

    const float4 v = *(const float4*)(x + (size_t)s * FDIM + f);
    float* a = agg + (size_t)d * FDIM + f;
    unsafeAtomicAdd(a + 0, v.x * w);
    unsafeAtomicAdd(a + 1, v.y * w);
    unsafeAtomicAdd(a + 2, v.z * w);
    unsafeAtomicAdd(a + 3, v.w * w);
}

// ------------------------------------------------- Tensor Data Mover helper
// 2D tile load Global -> LDS (D# per CDNA5 ISA ch.8):
//   group0: count=1 | lds_addr | 57b global_addr | type=2
//   group1: data_size=4B, pad cfg, tensor_dim0/1 (remaining extent from tile
//           origin -> OOB rows read as zero), tile_dim0/1, dim0 stride
//   groups 2/3: zero (2D tensor)
__device__ __forceinline__ void tdm_load_tile_2d(
    unsigned lds_byte_addr, const float* gptr,
    unsigned tensor_d0, unsigned tensor_d1,
    unsigned tile_d0, unsigned tile_d1,
    unsigned stride0_elems, unsigned pad_cfg) {
    unsigned long long ga = (unsigned long long)(uintptr_t)gptr;

    v4u g0;
    g0.x = 1u;                                              // count=1, user mode
    g0.y = lds_byte_addr;                                   // lds_addr
    g0.z = (unsigned)(ga & 0xFFFFFFFFu);                    // global_addr[31:0]
    g0.w = (unsigned)((ga >> 32) & 0x01FFFFFFu) | (2u << 30); // [56:32] | type=2

    v8i g1;
    g1[0] = (int)((2u << 16) | pad_cfg);   // wg_mask=0 | data_size=4B | pad cfg
    g1[1] = (int)((tensor_d0 & 0xFFFFu) << 16);             // abar=0 | td0.lo
    g1[2] = (int)(((tensor_d0 >> 16) & 0xFFFFu) | ((tensor_d1 & 0xFFFFu) << 16));
    g1[3] = (int)(((tensor_d1 >> 16) & 0xFFFFu) | ((tile_d0 & 0xFFFFu) << 16));
    g1[4] = (int)(tile_d1 & 0xFFFFu);                       // tile_dim2=0
    g1[5] = (int)stride0_elems;                             // dim0 stride lo32
    g1[6] = 0;                                              // stride hi, dim1 stride
    g1[7] = 0;

    v4i gz = {0, 0, 0, 0};
#if defined(__clang_major__) && (__clang_major__ >= 23)
    v8i gz8 = {0, 0, 0, 0, 0, 0, 0, 0};
    __builtin_amdgcn_tensor_load_to_lds(g0, g1, gz, gz, gz8, 0);
#else
    __builtin_amdgcn_tensor_load_to_lds(g0, g1, gz, gz, 0);
#endif
}

// pad_enable | pad_interval=4 (every 32 DWORDs = one A row) | pad_amount=0 (1 DW)
#define TDM_PAD_A ((1u << 20) | (4u << 22))

// ------------------------------------------------------------- WMMA GEMM
// out[i,j] = relu?( agg[i,:] @ W[:,j] + bias[j] )   (agg already norm-scaled)
// Block: 256 threads = 8 waves. Tile: 32 rows x 64 cols (wave -> 16x16 tile).
// Double-buffered LDS: TDM DMA of tile i+1 overlaps WMMA compute of tile i.
#define BM 32
#define BN 64
#define BK 32
#define LDA 33   // TDM hardware padding: 1 DWORD after each 32-DWORD row
#define LDB 64
#define KITERS (FDIM / BK)

__global__ __launch_bounds__(256)
void gcn_gemm_wmma_kernel(const float* __restrict__ A,    // [M, 512] = agg
                          const float* __restrict__ W,    // [512, 512]
                          const float* __restrict__ bias, // [512]
                          float* __restrict__ out,        // [M, 512]
                          int M, int relu) {
    __shared__ float sA[2][BM * LDA];
    __shared__ float sB[2][BK * LDB];

    const int tid = threadIdx.x;
    const int rowBase = blockIdx.y * BM;
    const int colBase = blockIdx.x * BN;

    const int wave = tid >> 5;
    const int lane = tid & 31;
    const int half = lane >> 4;       // 0: lanes 0-15, 1: lanes 16-31
    const int m    = lane & 15;
    const int wr = (wave >> 2) * 16;  // wave row offset within block tile
    const int wc = (wave & 3) * 16;   // wave col offset within block tile

    const unsigned remRows = (unsigned)(M - rowBase);
    const float* Abase = A + (size_t)rowBase * FDIM;
    const float* Wbase = W + colBase;

    v8f c = {};   // fp32 16x16 accumulator (8 VGPRs)

    // ---- prologue: stage tile 0 into buffer 0
    if (tid < 32) {
        tdm_load_tile_2d((unsigned)(uintptr_t)(&sA[0][0]), Abase,
                         FDIM, remRows, BK, BM, FDIM, TDM_PAD_A);
        tdm_load_tile_2d((unsigned)(uintptr_t)(&sB[0][0]), Wbase,
                         (unsigned)(FDIM - colBase), FDIM, BN, BK, FDIM, 0u);
    }

    for (int it = 0; it < KITERS; ++it) {
        const int cur = it & 1;
        if (tid < 32) {
            if (it + 1 < KITERS) {
                // issue next tile's DMAs into the other buffer, then wait for
                // the current buffer's two (older, in-order) DMAs only
                const int k1 = (it + 1) * BK;
                tdm_load_tile_2d((unsigned)(uintptr_t)(&sA[cur ^ 1][0]),
                                 Abase + k1,
                                 (unsigned)(FDIM - k1), remRows,
                                 BK, BM, FDIM, TDM_PAD_A);
                tdm_load_tile_2d((unsigned)(uintptr_t)(&sB[cur ^ 1][0]),
                                 Wbase + (size_t)k1 * FDIM,
                                 (unsigned)(FDIM - colBase), (unsigned)(FDIM - k1),
                                 BN, BK, FDIM, 0u);
                __builtin_amdgcn_s_wait_tensorcnt(2);
            } else {
                __builtin_amdgcn_s_wait_tensorcnt(0);
            }
        }
        __syncthreads();

        // ---- 8 chained fp32 WMMAs (K advances by 4 each)
        // A frag (16x4 f32): lanes 0-15 hold K=kk,kk+1; lanes 16-31 K=kk+2,kk+3
        // B frag (4x16 f32): VGPR0 rows K=kk / kk+2, VGPR1 rows K=kk+1 / kk+3
        const float* sAc = &sA[cur][0];
        const float* sBc = &sB[cur][0];
#pragma unroll
        for (int kk = 0; kk < BK; kk += 4) {
            v2f afrag, bfrag;
            const float* ap = sAc + (wr + m) * LDA + kk + 2 * half;
            afrag.x = ap[0];
            afrag.y = ap[1];
            const float* bp = sBc + (kk + 2 * half) * LDB + wc + m;
            bfrag.x = bp[0];
            bfrag.y = bp[LDB];
            c = __builtin_amdgcn_wmma_f32_16x16x4_f32(
                    /*neg_a=*/false, afrag,
                    /*neg_b=*/false, bfrag,
                    /*c_mod=*/(short)0, c,
                    /*reuse_a=*/false, /*reuse_b=*/false);
        }
        __syncthreads();   // reads of buffer `cur` done before it is re-filled
    }

    // ---- epilogue: bias + optional ReLU, store
    // C/D layout: VGPR v -> row (v + 8*half), col = m (within 16x16 tile)
    const int ocol = colBase + wc + m;
    const float bv = bias[ocol];
#pragma unroll
    for (int v = 0; v < 8; ++v) {
        int orow = rowBase + wr + v + 8 * half;
        if (orow < M) {
            float val = c[v] + bv;
            if (relu) val = fmaxf(val, 0.0f);
            out[(size_t)orow * FDIM + ocol] = val;
        }
    }
}

// ---------------------------------------------------------------- launcher
static inline void run_conv(const float* x, const float* W, const float* b,
                            const int* src, const int* dst,
                            const float* norm_out, const float* norm_in,
                            float* agg, float* out, int relu,
                            hipStream_t stream) {
    const int nfeat = N_NODES * FDIM;
    gcn_zero_kernel<<<4096, 256, 0, stream>>>(agg, nfeat);
    gcn_aggregate_kernel<<<N_EDGES / 2, 256, 0, stream>>>(x, src, dst,
                                                          norm_out, norm_in, agg);
    dim3 grid(FDIM / BN, (N_NODES + BM - 1) / BM);
    gcn_gemm_wmma_kernel<<<grid, 256, 0, stream>>>(agg, W, b, out, N_NODES, relu);
}

extern "C" void kernel_launch(void* const* d_in, const int* in_sizes, int n_in,
                              void* d_out, int out_size, void* d_ws, size_t ws_size,
                              hipStream_t stream) {
    const float* inputs = (const float*)d_in[0];
    const float* W1 = (const float*)d_in[1];
    const float* b1 = (const float*)d_in[2];
    const float* W2 = (const float*)d_in[3];
    const float* b2 = (const float*)d_in[4];
    const float* W3 = (const float*)d_in[5];
    const float* b3 = (const float*)d_in[6];
    const int* src = (const int*)d_in[7];
    const int* dst = (const int*)d_in[8];

    const size_t nodeF = (size_t)N_NODES * FDIM;
    float* out = (float*)d_out;
    float* h4 = out;              // first return value
    float* h3 = out + nodeF;      // second
    float* h2 = out + 2 * nodeF;  // third

    float* ws   = (float*)d_ws;
    float* agg  = ws;                 ws += nodeF;
    float* h1   = ws;                 ws += nodeF;
    float* deg  = ws;                 ws += 2 * N_NODES;  // [deg_out | deg_in]
    float* nrm  = ws;                 ws += 2 * N_NODES;  // [norm_out | norm_in]
    float* norm_out = nrm;
    float* norm_in  = nrm + N_NODES;

    // degrees + norms
    gcn_zero_kernel<<<64, 256, 0, stream>>>(deg, 2 * N_NODES);
    gcn_degree_kernel<<<(N_EDGES + 255) / 256, 256, 0, stream>>>(src, dst,
                                                                 deg, deg + N_NODES);
    gcn_norm_kernel<<<(2 * N_NODES + 255) / 256, 256, 0, stream>>>(deg, nrm,
                                                                   2 * N_NODES);

    // layer 1: inputs -> h1 (relu)
    run_conv(inputs, W1, b1, src, dst, norm_out, norm_in, agg, h1, 1, stream);
    // layer 2: h1 -> h2 (relu)
    run_conv(h1, W2, b2, src, dst, norm_out, norm_in, agg, h2, 1, stream);
    // layers 3 & 4 share the aggregation of h2: do it once, two GEMMs
    {
        const int nfeat = N_NODES * FDIM;
        gcn_zero_kernel<<<4096, 256, 0, stream>>>(agg, nfeat);
        gcn_aggregate_kernel<<<N_EDGES / 2, 256, 0, stream>>>(h2, src, dst,
                                                              norm_out, norm_in, agg);
        dim3 grid(FDIM / BN, (N_NODES + BM - 1) / BM);
        gcn_gemm_wmma_kernel<<<grid, 256, 0, stream>>>(agg, W2, b2, h3,
                                                       N_NODES, 1);
        gcn_gemm_wmma_kernel<<<grid, 256, 0, stream>>>(agg, W3, b3, h4,
                                                       N_NODES, 0);
    }
}